// ResNetSpherical_60026462929152
// MI455X (gfx1250) — compile-verified
//
#include <hip/hip_runtime.h>

typedef __attribute__((ext_vector_type(2))) float v2f;
typedef __attribute__((ext_vector_type(8))) float v8f;

static constexpr int Bn = 8;
static constexpr int Nn = 12288;
static constexpr int Cc = 64;
static constexpr int Ee = Nn * 20;             // 245760 edges
static constexpr int Mm = Nn * Bn;             // 98304 rows of the GEMM
static constexpr long long ELT = (long long)Mm * Cc;  // 6,291,456 elements

// ---------------------------------------------------------------------------
// Layout transposes: external [B,N,C] <-> working [N,B,C]  (float4 over C)
// ---------------------------------------------------------------------------
__global__ __launch_bounds__(256)
void transpose_in(const float4* __restrict__ x, float4* __restrict__ xt) {
  long long i = (long long)blockIdx.x * 256 + threadIdx.x;   // [N,B,C/4]
  int c4 = (int)(i & 15);
  int b  = (int)((i >> 4) & 7);
  long long n = i >> 7;
  xt[i] = x[(b * (long long)Nn + n) * 16 + c4];
}

__global__ __launch_bounds__(256)
void transpose_out(const float4* __restrict__ xt, float4* __restrict__ out) {
  long long i = (long long)blockIdx.x * 256 + threadIdx.x;   // [B,N,C/4]
  int c4 = (int)(i & 15);
  long long t = i >> 4;
  long long n = t % Nn;
  long long b = t / Nn;
  out[i] = xt[(n * Bn + b) * 16 + c4];
}

// ---------------------------------------------------------------------------
// dst = -src  (prepares x2 = 2*L@x1 - x0 via scaled atomic scatter)
// ---------------------------------------------------------------------------
__global__ __launch_bounds__(256)
void neg_copy(const float4* __restrict__ src, float4* __restrict__ dst) {
  long long i = (long long)blockIdx.x * 256 + threadIdx.x;
  float4 v = src[i];
  dst[i] = make_float4(-v.x, -v.y, -v.z, -v.w);
}

// ---------------------------------------------------------------------------
// SpMM scatter: y[rows[e], :, :] += scale * vals[e] * x[cols[e], :, :]
// one thread per (edge, float4-of-128); b128 gather + 4x global_atomic_add_f32
// ---------------------------------------------------------------------------
__global__ __launch_bounds__(256)
void spmm_scatter(const int* __restrict__ rows, const int* __restrict__ cols,
                  const float* __restrict__ vals, const float* __restrict__ x,
                  float* __restrict__ y, float scale) {
  long long i = (long long)blockIdx.x * 256 + threadIdx.x;   // [0, E*128)
  int e = (int)(i >> 7);
  int j = (int)(i & 127) << 2;
  float v = vals[e] * scale;
  float4 xv = *(const float4*)(x + (long long)cols[e] * 512 + j);
  float* yp = y + (long long)rows[e] * 512 + j;
  unsafeAtomicAdd(yp + 0, v * xv.x);
  unsafeAtomicAdd(yp + 1, v * xv.y);
  unsafeAtomicAdd(yp + 2, v * xv.z);
  unsafeAtomicAdd(yp + 3, v * xv.w);
}

// ---------------------------------------------------------------------------
// Chebyshev GEMM:  out[M,64] = A0@W[0] + A1@W[1] + A2@W[2]
//   M = 98304, K = 3*64 = 192, N = 64.  fp32 WMMA 16x16x4, 48 k-steps.
// Each wave owns one 16-row tile and ALL 64 output columns (4 accumulators):
// every A fragment feeds 4 independent WMMA chains -> 4x arithmetic intensity
// on the A stream and no A re-reads across an n-tile grid dimension.
// MODE 0: Out = acc                       (conv1, raw h for BN)
// MODE 1: Out = (acc + bias)*rz + Out     (conv2 + rezero residual, in place)
// ---------------------------------------------------------------------------
template<int MODE>
__global__ __launch_bounds__(256)
void gemm_cheb(const float* __restrict__ A0, const float* __restrict__ A1,
               const float* __restrict__ A2, const float* __restrict__ W,
               float* __restrict__ Out, const float* __restrict__ bias,
               const float* __restrict__ rz) {
  __shared__ float2 smB[48 * 32 * 4];      // 48 KB: B frags, 48 steps x 4 n-tiles
  const int tid  = threadIdx.x;
  const int lane = tid & 31;
  const int wave = tid >> 5;

  // Stage B fragments.  fp32 B 4x16 layout (mirrors A):
  //  lanes 0-15 carry {B[kc+0][n], B[kc+1][n]}, lanes 16-31 carry {kc+2, kc+3}.
  for (int p = tid; p < 48 * 32 * 4; p += 256) {
    int t  = p & 3;                                    // n-tile 0..3
    int l  = (p >> 2) & 31;                            // lane
    int s  = p >> 7;                                   // k-step 0..47
    int kk = s >> 4;                                   // which weight matrix
    int kc = ((s & 15) << 2) + ((l >> 4) << 1);        // k base within 0..63
    int n  = t * 16 + (l & 15);
    const float* w = W + kk * (Cc * Cc);
    smB[p] = make_float2(w[kc * Cc + n], w[(kc + 1) * Cc + n]);
  }
  __syncthreads();

  const int mtile = blockIdx.x * 8 + wave;             // 0..6143
  const int row   = mtile * 16 + (lane & 15);
  const int koff  = (lane >> 4) << 1;                  // 0 or 2 (A layout halves)
  const float* a0 = A0 + (long long)row * Cc + koff;
  const float* a1 = A1 + (long long)row * Cc + koff;
  const float* a2 = A2 + (long long)row * Cc + koff;

  v8f acc0 = {}, acc1 = {}, acc2 = {}, acc3 = {};
#pragma unroll
  for (int s = 0; s < 48; ++s) {
    const float* ap = (s < 16) ? a0 : (s < 32) ? a1 : a2;
    v2f a = *(const v2f*)(ap + ((s & 15) << 2));
    const float2* bp = &smB[(s * 32 + lane) * 4];
    v2f b0 = *(const v2f*)(bp + 0);
    v2f b1 = *(const v2f*)(bp + 1);
    v2f b2 = *(const v2f*)(bp + 2);
    v2f b3 = *(const v2f*)(bp + 3);
    acc0 = __builtin_amdgcn_wmma_f32_16x16x4_f32(false, a, false, b0, (short)0, acc0, false, false);
    acc1 = __builtin_amdgcn_wmma_f32_16x16x4_f32(false, a, false, b1, (short)0, acc1, false, false);
    acc2 = __builtin_amdgcn_wmma_f32_16x16x4_f32(false, a, false, b2, (short)0, acc2, false, false);
    acc3 = __builtin_amdgcn_wmma_f32_16x16x4_f32(false, a, false, b3, (short)0, acc3, false, false);
  }

  // D layout: vgpr v, lanes 0-15 -> row v, lanes 16-31 -> row v+8; col = lane&15
  const int oc    = lane & 15;
  const int orow0 = mtile * 16 + ((lane >> 4) << 3);
  const float r  = (MODE == 1) ? rz[0] : 0.0f;
#pragma unroll
  for (int t = 0; t < 4; ++t) {
    v8f acc = (t == 0) ? acc0 : (t == 1) ? acc1 : (t == 2) ? acc2 : acc3;
    const int ocol = t * 16 + oc;
    if (MODE == 0) {
#pragma unroll
      for (int v = 0; v < 8; ++v)
        Out[(long long)(orow0 + v) * Cc + ocol] = acc[v];
    } else {
      const float bz = bias[ocol];
#pragma unroll
      for (int v = 0; v < 8; ++v) {
        long long idx = (long long)(orow0 + v) * Cc + ocol;
        Out[idx] = (acc[v] + bz) * r + Out[idx];
      }
    }
  }
}

// ---------------------------------------------------------------------------
// BatchNorm: per-channel sum / sumsq over all 98304 rows, then scale/shift.
// stats layout: [0:64)=sum  [64:128)=sumsq  [128:192)=scale  [192:256)=shift
// ---------------------------------------------------------------------------
__global__ __launch_bounds__(256)
void bn_stats(const float* __restrict__ h, float* __restrict__ stats) {
  __shared__ float s1[256];
  __shared__ float s2[256];
  const int tid = threadIdx.x;
  const int c = tid & 63;
  const int q = tid >> 6;
  float sum = 0.f, sq = 0.f;
  for (int r = blockIdx.x * 4 + q; r < Mm; r += gridDim.x * 4) {
    float v = h[(long long)r * Cc + c];
    sum += v;
    sq  += v * v;
  }
  s1[tid] = sum;
  s2[tid] = sq;
  __syncthreads();
  if (tid < 64) {
    float a = s1[tid] + s1[tid + 64] + s1[tid + 128] + s1[tid + 192];
    float b = s2[tid] + s2[tid + 64] + s2[tid + 128] + s2[tid + 192];
    unsafeAtomicAdd(&stats[c], a);
    unsafeAtomicAdd(&stats[64 + c], b);
  }
}

__global__ void bn_finalize(float* __restrict__ stats, const float* __restrict__ g,
                            const float* __restrict__ b) {
  int c = threadIdx.x;
  if (c < 64) {
    const float inv = 1.0f / (float)Mm;
    float mean = stats[c] * inv;
    float var  = stats[64 + c] * inv - mean * mean;
    float sc   = g[c] * rsqrtf(var + 1e-5f);
    stats[128 + c] = sc;
    stats[192 + c] = b[c] - mean * sc;
  }
}

__global__ __launch_bounds__(256)
void bn_apply_relu(float4* __restrict__ h, const float* __restrict__ stats) {
  long long i = (long long)blockIdx.x * 256 + threadIdx.x;   // [0, ELT/4)
  int c4 = (int)(i & 15) << 2;
  float4 sc = *(const float4*)(stats + 128 + c4);
  float4 sh = *(const float4*)(stats + 192 + c4);
  float4 v = h[i];
  v.x = fmaxf(v.x * sc.x + sh.x, 0.f);
  v.y = fmaxf(v.y * sc.y + sh.y, 0.f);
  v.z = fmaxf(v.z * sc.z + sh.z, 0.f);
  v.w = fmaxf(v.w * sc.w + sh.w, 0.f);
  h[i] = v;
}

// ---------------------------------------------------------------------------
// Host-side sequencing
// ---------------------------------------------------------------------------
static void run_resblock(float* xt, float* x1, float* x2, float* h, float* st,
                         const int* rows, const int* cols, const float* vals,
                         const float* w1, const float* g, const float* bb,
                         const float* w2, const float* bias2, const float* rz,
                         hipStream_t stream) {
  const size_t Sb = (size_t)Mm * Cc * sizeof(float);
  const int ELT4_GRID = (int)(ELT / 4 / 256);                 // 6144
  const int SP_GRID   = (int)(((long long)Ee * 128) / 256);   // 122880
  const dim3 GG(Mm / 16 / 8);                                 // 768 blocks

  // x1 = L @ xt
  hipMemsetAsync(x1, 0, Sb, stream);
  spmm_scatter<<<SP_GRID, 256, 0, stream>>>(rows, cols, vals, xt, x1, 1.0f);
  // x2 = 2 * L @ x1 - xt
  neg_copy<<<ELT4_GRID, 256, 0, stream>>>((const float4*)xt, (float4*)x2);
  spmm_scatter<<<SP_GRID, 256, 0, stream>>>(rows, cols, vals, x1, x2, 2.0f);
  // h = xt@w1[0] + x1@w1[1] + x2@w1[2]
  gemm_cheb<0><<<GG, 256, 0, stream>>>(xt, x1, x2, w1, h, nullptr, nullptr);
  // BN + ReLU on h
  hipMemsetAsync(st, 0, 256 * sizeof(float), stream);
  bn_stats<<<1024, 256, 0, stream>>>(h, st);
  bn_finalize<<<1, 64, 0, stream>>>(st, g, bb);
  bn_apply_relu<<<ELT4_GRID, 256, 0, stream>>>((float4*)h, st);
  // second Chebyshev conv on h
  hipMemsetAsync(x1, 0, Sb, stream);
  spmm_scatter<<<SP_GRID, 256, 0, stream>>>(rows, cols, vals, h, x1, 1.0f);
  neg_copy<<<ELT4_GRID, 256, 0, stream>>>((const float4*)h, (float4*)x2);
  spmm_scatter<<<SP_GRID, 256, 0, stream>>>(rows, cols, vals, x1, x2, 2.0f);
  // xt = (h@w2[0] + x1@w2[1] + x2@w2[2] + bias2) * rz + xt   (in place)
  gemm_cheb<1><<<GG, 256, 0, stream>>>(h, x1, x2, w2, xt, bias2, rz);
}

extern "C" void kernel_launch(void* const* d_in, const int* in_sizes, int n_in,
                              void* d_out, int out_size, void* d_ws, size_t ws_size,
                              hipStream_t stream) {
  const float* x    = (const float*)d_in[0];
  const int*   rows = (const int*)d_in[1];
  const int*   cols = (const int*)d_in[2];
  const float* vals = (const float*)d_in[3];
  const float* w1_1 = (const float*)d_in[4];
  const float* g1_1 = (const float*)d_in[5];
  const float* b1_1 = (const float*)d_in[6];
  const float* w1_2 = (const float*)d_in[7];
  const float* bi12 = (const float*)d_in[8];
  const float* rz1  = (const float*)d_in[9];
  const float* w2_1 = (const float*)d_in[10];
  const float* g2_1 = (const float*)d_in[11];
  const float* b2_1 = (const float*)d_in[12];
  const float* w2_2 = (const float*)d_in[13];
  const float* bi22 = (const float*)d_in[14];
  const float* rz2  = (const float*)d_in[15];

  const size_t S = (size_t)Mm * Cc;      // floats per [N,B,C] buffer
  float* xt = (float*)d_ws;
  float* x1 = xt + S;
  float* x2 = x1 + S;
  float* h  = x2 + S;
  float* st = h + S;                     // 256 floats of BN stats

  const int ELT4_GRID = (int)(ELT / 4 / 256);
  transpose_in<<<ELT4_GRID, 256, 0, stream>>>((const float4*)x, (float4*)xt);

  run_resblock(xt, x1, x2, h, st, rows, cols, vals,
               w1_1, g1_1, b1_1, w1_2, bi12, rz1, stream);
  run_resblock(xt, x1, x2, h, st, rows, cols, vals,
               w2_1, g2_1, b2_1, w2_2, bi22, rz2, stream);

  transpose_out<<<ELT4_GRID, 256, 0, stream>>>((const float4*)xt, (float4*)d_out);
}